// MySoftmaxQAT_61675730371184
// MI455X (gfx1250) — compile-verified
//
#include <hip/hip_runtime.h>

// ---------------------------------------------------------------------------
// MySoftmaxQAT on gfx1250: fused fake-quant + exp-LUT + row softmax normalize.
// Memory-bound (1.07 GB traffic, ~46us floor @ 23.3 TB/s). Strategy:
//   - closed-form LUT:  table[q] = exp((q - zp) * s)  ->  v_exp_f32
//   - persistent blocks, one 8KB row per iteration, double-buffered into LDS
//     via CDNA5 async loads (GLOBAL_LOAD_ASYNC_TO_LDS_B128 + S_WAIT_ASYNCCNT)
//   - wave32 shuffle reduction + 8 LDS partials for the row sum
//   - non-temporal 128-bit output stores (don't pollute L2)
// ---------------------------------------------------------------------------

#define ROWLEN 2048
#define TPB    256
#define EPT    (ROWLEN / TPB)   // 8 floats per thread
#define NWAVES (TPB / 32)       // 8 wave32s per block
#define LOG2E  1.4426950408889634f

typedef int   v4i __attribute__((ext_vector_type(4)));
typedef float v4f __attribute__((ext_vector_type(4)));

#if defined(__AMDGCN__)
typedef __attribute__((address_space(1))) v4i* gv4i_p;   // global
typedef __attribute__((address_space(3))) v4i* lv4i_p;   // LDS
#endif

// --- CDNA5 async copy: 16 bytes global -> LDS, tracked by ASYNCcnt ---------
__device__ __forceinline__ void async_copy16(const float* g, float* l) {
#if defined(__AMDGCN__)
#if __has_builtin(__builtin_amdgcn_global_load_async_to_lds_b128)
    __builtin_amdgcn_global_load_async_to_lds_b128(
        (gv4i_p)(void*)g, (lv4i_p)(void*)l, /*offset=*/0, /*cpol=*/0);
#else
    unsigned loff = (unsigned)(unsigned long long)
        (__attribute__((address_space(3))) char*)l;
    asm volatile("global_load_async_to_lds_b128 %0, %1, off"
                 :: "v"(loff), "v"(g)
                 : "memory");
#endif
#else
    (void)g; (void)l;   // host pass: never executed
#endif
}

__device__ __forceinline__ void wait_async0() {
#if defined(__AMDGCN__)
#if __has_builtin(__builtin_amdgcn_s_wait_asynccnt)
    __builtin_amdgcn_s_wait_asynccnt(0);
#else
    asm volatile("s_wait_asynccnt 0" ::: "memory");
#endif
#endif
}

__device__ __forceinline__ float fast_exp2(float x) {
#if defined(__AMDGCN__) && __has_builtin(__builtin_amdgcn_exp2f)
    return __builtin_amdgcn_exp2f(x);
#else
    return x;           // host pass stub: never executed
#endif
}

__device__ __forceinline__ float fast_rcp(float x) {
#if defined(__AMDGCN__) && __has_builtin(__builtin_amdgcn_rcpf)
    return __builtin_amdgcn_rcpf(x);
#else
    return 1.0f / x;
#endif
}

__global__ __launch_bounds__(TPB) void softmax_qat_kernel(
    const float* __restrict__ X,
    const float* __restrict__ scale,
    const float* __restrict__ zero_point,
    float* __restrict__ Y,
    int nrows)
{
    __shared__ float buf[2][ROWLEN];     // double-buffered row staging (16 KB)
    __shared__ float partials[NWAVES];

    const int tid = threadIdx.x;

    const float s     = scale[0];
    const float zp    = zero_point[0];
    const float inv_s = 1.0f / s;
    const float c1    = s * LOG2E;       // exp((q-zp)*s) = exp2(q*c1 + c0)
    const float c0    = -zp * c1;

    const long stride = (long)gridDim.x;
    long r = (long)blockIdx.x;

    // Preamble prefetch of the first row owned by this block.
    if (r < nrows) {
        const float* src = X + (size_t)r * ROWLEN;
        async_copy16(src + tid * 4,        &buf[0][tid * 4]);
        async_copy16(src + tid * 4 + 1024, &buf[0][tid * 4 + 1024]);
    }

    int cur = 0;
    for (; r < nrows; r += stride) {
        // Wait for this wave's async loads of buf[cur]; barrier makes the
        // whole buffer visible to all 8 waves (and guarantees the previous
        // row's buffer is fully consumed before we overwrite it below).
        wait_async0();
        __syncthreads();

        // Prefetch next row into the other buffer (overlaps with compute).
        long rn = r + stride;
        if (rn < nrows) {
            const float* src = X + (size_t)rn * ROWLEN;
            async_copy16(src + tid * 4,        &buf[cur ^ 1][tid * 4]);
            async_copy16(src + tid * 4 + 1024, &buf[cur ^ 1][tid * 4 + 1024]);
        }

        // ---- compute: 8 contiguous floats per thread --------------------
        const v4f a = *(const v4f*)&buf[cur][tid * EPT];
        const v4f b = *(const v4f*)&buf[cur][tid * EPT + 4];
        float xs[EPT] = {a.x, a.y, a.z, a.w, b.x, b.y, b.z, b.w};

        float ys[EPT];
        float lsum = 0.0f;
#pragma unroll
        for (int i = 0; i < EPT; ++i) {
            // fake-quantize: q = rint(clamp(x/s + zp, 0, 255))
            float q = __builtin_rintf(
                fminf(fmaxf(fmaf(xs[i], inv_s, zp), 0.0f), 255.0f));
            // LUT gather == closed form: exp((q - zp) * s)
            float e = fast_exp2(fmaf(q, c1, c0));
            ys[i] = e;
            lsum += e;
        }

        // ---- row-sum reduction: wave32 shuffle + LDS partials -----------
#pragma unroll
        for (int m = 16; m >= 1; m >>= 1)
            lsum += __shfl_xor(lsum, m, 32);
        if ((tid & 31) == 0)
            partials[tid >> 5] = lsum;
        __syncthreads();

        float tot = 0.0f;
#pragma unroll
        for (int w = 0; w < NWAVES; ++w)
            tot += partials[w];
        const float rinv = fast_rcp(tot);

        // ---- normalized output, non-temporal 128-bit stores -------------
        v4f o0 = {ys[0] * rinv, ys[1] * rinv, ys[2] * rinv, ys[3] * rinv};
        v4f o1 = {ys[4] * rinv, ys[5] * rinv, ys[6] * rinv, ys[7] * rinv};
        v4f* dst = (v4f*)(Y + (size_t)r * ROWLEN + tid * EPT);
#if __has_builtin(__builtin_nontemporal_store)
        __builtin_nontemporal_store(o0, dst);
        __builtin_nontemporal_store(o1, dst + 1);
#else
        dst[0] = o0;
        dst[1] = o1;
#endif
        cur ^= 1;
    }
}

extern "C" void kernel_launch(void* const* d_in, const int* in_sizes, int n_in,
                              void* d_out, int out_size, void* d_ws, size_t ws_size,
                              hipStream_t stream) {
    const float* X  = (const float*)d_in[0];
    const float* s  = (const float*)d_in[1];
    const float* zp = (const float*)d_in[2];
    float* Y = (float*)d_out;

    const int nrows = in_sizes[0] / ROWLEN;     // 65536 for (2,16,2048,2048)
    int grid = nrows < 8192 ? nrows : 8192;     // persistent: 8 rows per block
    if (grid < 1) grid = 1;

    softmax_qat_kernel<<<grid, TPB, 0, stream>>>(X, s, zp, Y, nrows);
}